// ProtDiGCNEncoderDecoder_ngram_59889023976011
// MI455X (gfx1250) — compile-verified
//
#include <hip/hip_runtime.h>
#include <math.h>

#define N_NODES 50000
#define E_EDGES 640000
#define F_INDIM 96
#define H_DIM   128
#define NCLS    256
#define SCAN_BLOCKS ((N_NODES + 255) / 256)   // 196

#define ROWS_PER_BLOCK 80                     // 50000 = 625 * 80 exactly
#define TILES_PER_WAVE 5                      // 80 / 16
#define LDS_STRIDE_MAX 132                    // K(max 128) + 4 pad dwords

typedef float v2f __attribute__((ext_vector_type(2)));
typedef float v8f __attribute__((ext_vector_type(8)));

// ---------------------------------------------------------------------------
// positional encoding: xpe[i,j] = x[i,j] + pe_flat[j % 96]
// ---------------------------------------------------------------------------
__global__ void pe_add_kernel(const float* __restrict__ x,
                              const float* __restrict__ pe,
                              float* __restrict__ xpe) {
    int idx = blockIdx.x * blockDim.x + threadIdx.x;
    if (idx < N_NODES * F_INDIM) {
        xpe[idx] = x[idx] + pe[idx % F_INDIM];
    }
}

// ---------------------------------------------------------------------------
// fold skip weights into main weights (prop is linear, shared edge set)
// ---------------------------------------------------------------------------
__global__ void combine_wb_kernel(const float* __restrict__ Wmi, const float* __restrict__ Wmo,
                                  const float* __restrict__ Wsk,
                                  const float* __restrict__ bmi, const float* __restrict__ bmo,
                                  const float* __restrict__ bsi, const float* __restrict__ bso,
                                  float* __restrict__ WcIn, float* __restrict__ WcOut,
                                  float* __restrict__ bcIn, float* __restrict__ bcOut,
                                  int kd) {
    int idx = blockIdx.x * blockDim.x + threadIdx.x;
    if (idx < kd) {
        float s = Wsk[idx];
        WcIn[idx]  = Wmi[idx] + s;
        WcOut[idx] = Wmo[idx] + s;
    }
    if (idx < H_DIM) {
        bcIn[idx]  = bmi[idx] + bsi[idx];
        bcOut[idx] = bmo[idx] + bso[idx];
    }
}

// ---------------------------------------------------------------------------
// WMMA f32 GEMM:  C[N,D] = A[N,K] @ W[K,D]  (+ optional bias)
// block = 8 waves, owns 80 rows staged in LDS; each wave: 16 cols,
// 5 accumulator tiles, 5 x V_WMMA_F32_16X16X4_F32 per K-step of 4.
// ---------------------------------------------------------------------------
__global__ void gemm_wmma_f32_kernel(const float* __restrict__ A,
                                     const float* __restrict__ W,
                                     const float* __restrict__ bias,
                                     float* __restrict__ C,
                                     int K, int D) {
    __shared__ float As[ROWS_PER_BLOCK * LDS_STRIDE_MAX];
    const int ldk  = K + 4;                       // padded row stride (dwords)
    const int tid  = threadIdx.x;
    const int wave = tid >> 5;
    const int lane = tid & 31;
    const int row0 = blockIdx.x * ROWS_PER_BLOCK;
    const int col0 = (blockIdx.y * 8 + wave) * 16;

    // cooperative stage of A[row0 .. row0+79][0..K) into LDS (float4 granular)
    const int k4 = K >> 2;
    const int n4 = ROWS_PER_BLOCK * k4;
    for (int i4 = tid; i4 < n4; i4 += 256) {
        int r = i4 / k4;
        int c = (i4 - r * k4) << 2;
        const float4 v = *(const float4*)(A + (size_t)(row0 + r) * K + c);
        *(float4*)(&As[r * ldk + c]) = v;
    }
    __syncthreads();

    const int m    = lane & 15;
    const int koff = (lane >> 4) << 1;            // 0 / 2

    v8f acc[TILES_PER_WAVE];
#pragma unroll
    for (int t = 0; t < TILES_PER_WAVE; ++t)
#pragma unroll
        for (int e = 0; e < 8; ++e) acc[t][e] = 0.0f;

    for (int k0 = 0; k0 < K; k0 += 4) {
        v2f b;
        b[0] = W[(size_t)(k0 + koff) * D + col0 + m];
        b[1] = W[(size_t)(k0 + koff + 1) * D + col0 + m];
#pragma unroll
        for (int t = 0; t < TILES_PER_WAVE; ++t) {
            const v2f a = *(const v2f*)(&As[(t * 16 + m) * ldk + k0 + koff]);
            acc[t] = __builtin_amdgcn_wmma_f32_16x16x4_f32(
                false, a, false, b, (short)0, acc[t], false, false);
        }
    }

    const int col  = col0 + m;
    const float bv = bias ? bias[col] : 0.0f;
    const int half = (lane >> 4) << 3;            // 0 / 8
#pragma unroll
    for (int t = 0; t < TILES_PER_WAVE; ++t) {
        const int rbase = row0 + t * 16 + half;
#pragma unroll
        for (int r = 0; r < 8; ++r) {
            C[(size_t)(rbase + r) * D + col] = acc[t][r] + bv;
        }
    }
}

// ---------------------------------------------------------------------------
// CSR build: count / scan / finalize / fill   (once per direction per call)
// ---------------------------------------------------------------------------
__global__ void zero_int_kernel(int* __restrict__ p, int n) {
    int i = blockIdx.x * blockDim.x + threadIdx.x;
    if (i < n) p[i] = 0;
}

__global__ void count_kernel(const int* __restrict__ ei, int* __restrict__ cnt) {
    int e = blockIdx.x * blockDim.x + threadIdx.x;
    if (e < E_EDGES) atomicAdd(&cnt[ei[E_EDGES + e]], 1);
}

// block-local inclusive scan (Hillis-Steele in LDS), emit block totals
__global__ void scan_block_kernel(const int* __restrict__ cnt,
                                  int* __restrict__ incl, int* __restrict__ bsum) {
    __shared__ int tmp[256];
    int i = blockIdx.x * 256 + threadIdx.x;
    int v = (i < N_NODES) ? cnt[i] : 0;
    tmp[threadIdx.x] = v;
    __syncthreads();
    for (int off = 1; off < 256; off <<= 1) {
        int t = (threadIdx.x >= off) ? tmp[threadIdx.x - off] : 0;
        __syncthreads();
        tmp[threadIdx.x] += t;
        __syncthreads();
    }
    if (i < N_NODES) incl[i] = tmp[threadIdx.x];
    if (threadIdx.x == 255) bsum[blockIdx.x] = tmp[255];
}

// single-block exclusive scan of the block totals (nb <= 256)
__global__ void scan_top_kernel(int* __restrict__ bsum, int nb) {
    __shared__ int tmp[256];
    int v = (threadIdx.x < nb) ? bsum[threadIdx.x] : 0;
    tmp[threadIdx.x] = v;
    __syncthreads();
    for (int off = 1; off < 256; off <<= 1) {
        int t = (threadIdx.x >= off) ? tmp[threadIdx.x - off] : 0;
        __syncthreads();
        tmp[threadIdx.x] += t;
        __syncthreads();
    }
    if (threadIdx.x < nb) bsum[threadIdx.x] = tmp[threadIdx.x] - v;
}

// rowptr[i] = block-local-exclusive + block offset ; also copy into cursor
__global__ void finalize_rowptr_kernel(const int* __restrict__ incl,
                                       const int* __restrict__ cnt,
                                       const int* __restrict__ bsum,
                                       int* __restrict__ rowptr,
                                       int* __restrict__ cursor) {
    int i = blockIdx.x * 256 + threadIdx.x;
    if (i < N_NODES) {
        int r = incl[i] - cnt[i] + bsum[blockIdx.x];
        rowptr[i] = r;
        cursor[i] = r;
    }
    if (i == 0) rowptr[N_NODES] = E_EDGES;
}

__global__ void csr_fill_kernel(const int* __restrict__ ei, const float* __restrict__ ew,
                                int* __restrict__ cursor,
                                int* __restrict__ cs, float* __restrict__ cw) {
    int e = blockIdx.x * blockDim.x + threadIdx.x;
    if (e < E_EDGES) {
        int d = ei[E_EDGES + e];
        int pos = atomicAdd(&cursor[d], 1);
        cs[pos] = ei[e];
        cw[pos] = ew[e];
    }
}

// ---------------------------------------------------------------------------
// atomic-free gather: acc = sum_j w[j] * Y[src[j], lane*4..+3]
// edge indices/weights broadcast 32-at-a-time via lane shuffles
// ---------------------------------------------------------------------------
__device__ inline float4 gather_row(const float* __restrict__ Y,
                                    const int* __restrict__ cs,
                                    const float* __restrict__ cw,
                                    int start, int end, int lane) {
    float4 acc = make_float4(0.f, 0.f, 0.f, 0.f);
    for (int j = start; j < end; j += 32) {
        int idx = j + lane;
        int sv = 0; float wv = 0.f;
        if (idx < end) { sv = cs[idx]; wv = cw[idx]; }
        int cnt = min(32, end - j);
        for (int t = 0; t < cnt; ++t) {
            int s   = __shfl(sv, t, 32);
            float w = __shfl(wv, t, 32);
            const float4 y = *(const float4*)(Y + (size_t)s * H_DIM + lane * 4);
            acc.x += w * y.x; acc.y += w * y.y;
            acc.z += w * y.z; acc.w += w * y.w;
        }
    }
    return acc;
}

// fused per-layer: both-direction gathers + bias + C coeffs + residual + ReLU
__global__ void layer_gather_combine_kernel(
        const float* __restrict__ Yin,  const float* __restrict__ Yout,
        const int* __restrict__ rpin,  const int* __restrict__ csin,  const float* __restrict__ cwin,
        const int* __restrict__ rpout, const int* __restrict__ csout, const float* __restrict__ cwout,
        const float* __restrict__ bci, const float* __restrict__ bco,
        const float* __restrict__ Ci,  const float* __restrict__ Co,
        const float* __restrict__ res, float* __restrict__ H, int relu) {
    int node = (blockIdx.x * blockDim.x + threadIdx.x) >> 5;
    int lane = threadIdx.x & 31;
    if (node >= N_NODES) return;

    float4 ain  = gather_row(Yin,  csin,  cwin,  rpin[node],  rpin[node + 1],  lane);
    float4 aout = gather_row(Yout, csout, cwout, rpout[node], rpout[node + 1], lane);

    const int c4 = lane * 4;
    const float ci = Ci[node], co = Co[node];
    const float4 bi = *(const float4*)(bci + c4);
    const float4 bo = *(const float4*)(bco + c4);
    const float4 r  = *(const float4*)(res + (size_t)node * H_DIM + c4);

    float4 o;
    o.x = ci * (ain.x + bi.x) + co * (aout.x + bo.x) + r.x;
    o.y = ci * (ain.y + bi.y) + co * (aout.y + bo.y) + r.y;
    o.z = ci * (ain.z + bi.z) + co * (aout.z + bo.z) + r.z;
    o.w = ci * (ain.w + bi.w) + co * (aout.w + bo.w) + r.w;
    if (relu) {
        o.x = fmaxf(o.x, 0.f); o.y = fmaxf(o.y, 0.f);
        o.z = fmaxf(o.z, 0.f); o.w = fmaxf(o.w, 0.f);
    }
    *(float4*)(H + (size_t)node * H_DIM + c4) = o;
}

// ---------------------------------------------------------------------------
// in-place L2 normalize rows (wave per row, shuffle reduction)
// ---------------------------------------------------------------------------
__global__ void l2norm_kernel(float* __restrict__ emb) {
    int row  = (blockIdx.x * blockDim.x + threadIdx.x) >> 5;
    int lane = threadIdx.x & 31;
    if (row >= N_NODES) return;
    const float4 v = *(const float4*)(emb + (size_t)row * H_DIM + lane * 4);
    float ss = v.x * v.x + v.y * v.y + v.z * v.z + v.w * v.w;
#pragma unroll
    for (int off = 16; off; off >>= 1) ss += __shfl_xor(ss, off, 32);
    float inv = 1.0f / fmaxf(sqrtf(ss), 1e-12f);
    float4 o;
    o.x = v.x * inv; o.y = v.y * inv; o.z = v.z * inv; o.w = v.w * inv;
    *(float4*)(emb + (size_t)row * H_DIM + lane * 4) = o;
}

// ---------------------------------------------------------------------------
// in-place row log_softmax over 256 classes (block per row)
// ---------------------------------------------------------------------------
__global__ void logsoftmax_kernel(float* __restrict__ logits) {
    __shared__ float redmax[8];
    __shared__ float redsum[8];
    int row = blockIdx.x;
    int j   = threadIdx.x;
    float v = logits[(size_t)row * NCLS + j];

    float m = v;
#pragma unroll
    for (int off = 16; off; off >>= 1) m = fmaxf(m, __shfl_xor(m, off, 32));
    if ((threadIdx.x & 31) == 0) redmax[threadIdx.x >> 5] = m;
    __syncthreads();
    float mm = redmax[0];
#pragma unroll
    for (int t = 1; t < 8; ++t) mm = fmaxf(mm, redmax[t]);

    float e = __expf(v - mm);
    float s = e;
#pragma unroll
    for (int off = 16; off; off >>= 1) s += __shfl_xor(s, off, 32);
    if ((threadIdx.x & 31) == 0) redsum[threadIdx.x >> 5] = s;
    __syncthreads();
    float ss = 0.0f;
#pragma unroll
    for (int t = 0; t < 8; ++t) ss += redsum[t];

    logits[(size_t)row * NCLS + j] = v - mm - __logf(ss);
}

// ---------------------------------------------------------------------------
// orchestration
// ---------------------------------------------------------------------------
extern "C" void kernel_launch(void* const* d_in, const int* in_sizes, int n_in,
                              void* d_out, int out_size, void* d_ws, size_t ws_size,
                              hipStream_t stream) {
    (void)in_sizes; (void)n_in; (void)out_size; (void)ws_size;

    const float* x      = (const float*)d_in[0];
    const int*   ei_in  = (const int*)d_in[1];
    const float* ew_in  = (const float*)d_in[2];
    const int*   ei_out = (const int*)d_in[3];
    const float* ew_out = (const float*)d_in[4];
    const float* pe     = (const float*)d_in[5];

    const float *Wmi[3], *Wmo[3], *Wsk[3], *bmi[3], *bmo[3], *bsi[3], *bso[3], *Ci[3], *Co[3];
    for (int l = 0; l < 3; ++l) {
        int b = 6 + l * 9;
        Wmi[l] = (const float*)d_in[b + 0];
        Wmo[l] = (const float*)d_in[b + 1];
        Wsk[l] = (const float*)d_in[b + 2];
        bmi[l] = (const float*)d_in[b + 3];
        bmo[l] = (const float*)d_in[b + 4];
        bsi[l] = (const float*)d_in[b + 5];
        bso[l] = (const float*)d_in[b + 6];
        Ci[l]  = (const float*)d_in[b + 7];
        Co[l]  = (const float*)d_in[b + 8];
    }
    const float* res1_W = (const float*)d_in[33];
    const float* res1_b = (const float*)d_in[34];
    const float* dec_W  = (const float*)d_in[35];
    const float* dec_b  = (const float*)d_in[36];

    // ---- workspace layout (4-byte elements) ----
    float* ws = (float*)d_ws;
    const size_t NH = (size_t)N_NODES * H_DIM;
    float* XPE   = ws;                                  // N*96
    float* YIN   = XPE + (size_t)N_NODES * F_INDIM;     // N*128
    float* YOUT  = YIN  + NH;                           // N*128
    float* H1    = YOUT + NH;                           // N*128
    float* H2    = H1   + NH;                           // N*128 (RRES alias in layer0)
    float* WCIN  = H2   + NH;                           // 128*128
    float* WCOUT = WCIN + 16384;
    float* BCIN  = WCOUT + 16384;
    float* BCOUT = BCIN + 128;
    float* CWIN  = BCOUT + 128;                         // E floats
    float* CWOUT = CWIN + E_EDGES;                      // E floats
    int*   RPIN  = (int*)(CWOUT + E_EDGES);             // N+1
    int*   RPOUT = RPIN  + (N_NODES + 1);               // N+1
    int*   CSIN  = RPOUT + (N_NODES + 1);               // E
    int*   CSOUT = CSIN  + E_EDGES;                     // E
    int*   CNT   = CSOUT + E_EDGES;                     // N
    int*   CUR   = CNT   + N_NODES;                     // N
    int*   TINCL = CUR   + N_NODES;                     // N
    int*   BSUM  = TINCL + N_NODES;                     // 256

    float* logits = (float*)d_out;                              // N*256
    float* emb    = (float*)d_out + (size_t)N_NODES * NCLS;     // N*128 (H3 lives here)

    const int gemmBlocks = N_NODES / ROWS_PER_BLOCK;    // 625 exact
    const dim3 blk(256);
    const int nBlkN  = (N_NODES + 255) / 256;           // 196
    const int nBlkE  = (E_EDGES + 255) / 256;
    const int nBlkNW = (N_NODES * 32 + 255) / 256;      // wave-per-node grids

    // ---- CSR build, both directions (edge structure is reused by all layers)
    const int*   eis[2] = {ei_in, ei_out};
    const float* ews[2] = {ew_in, ew_out};
    int*   rps[2] = {RPIN, RPOUT};
    int*   css[2] = {CSIN, CSOUT};
    float* cws[2] = {CWIN, CWOUT};
    for (int d = 0; d < 2; ++d) {
        zero_int_kernel<<<nBlkN, blk, 0, stream>>>(CNT, N_NODES);
        count_kernel<<<nBlkE, blk, 0, stream>>>(eis[d], CNT);
        scan_block_kernel<<<SCAN_BLOCKS, blk, 0, stream>>>(CNT, TINCL, BSUM);
        scan_top_kernel<<<1, blk, 0, stream>>>(BSUM, SCAN_BLOCKS);
        finalize_rowptr_kernel<<<SCAN_BLOCKS, blk, 0, stream>>>(TINCL, CNT, BSUM, rps[d], CUR);
        csr_fill_kernel<<<nBlkE, blk, 0, stream>>>(eis[d], ews[d], CUR, css[d], cws[d]);
    }

    // ---- positional encoding
    pe_add_kernel<<<(N_NODES * F_INDIM + 255) / 256, blk, 0, stream>>>(x, pe, XPE);

    // ---- three DiGCN layers
    const float* hin = XPE;
    int K = F_INDIM;
    float* Hout[3] = {H1, H2, emb};      // layer3 result straight into d_out emb region
    const int reluFlag[3] = {1, 1, 0};

    for (int l = 0; l < 3; ++l) {
        combine_wb_kernel<<<(K * H_DIM + 255) / 256, blk, 0, stream>>>(
            Wmi[l], Wmo[l], Wsk[l], bmi[l], bmo[l], bsi[l], bso[l],
            WCIN, WCOUT, BCIN, BCOUT, K * H_DIM);

        gemm_wmma_f32_kernel<<<dim3(gemmBlocks, 1), blk, 0, stream>>>(hin, WCIN,  nullptr, YIN,  K, H_DIM);
        gemm_wmma_f32_kernel<<<dim3(gemmBlocks, 1), blk, 0, stream>>>(hin, WCOUT, nullptr, YOUT, K, H_DIM);

        const float* res;
        if (l == 0) {
            // res1 projection into H2 (free until layer1 writes it)
            gemm_wmma_f32_kernel<<<dim3(gemmBlocks, 1), blk, 0, stream>>>(XPE, res1_W, res1_b, H2, F_INDIM, H_DIM);
            res = H2;
        } else {
            res = (l == 1) ? H1 : H2;
        }

        layer_gather_combine_kernel<<<nBlkNW, blk, 0, stream>>>(
            YIN, YOUT, RPIN, CSIN, CWIN, RPOUT, CSOUT, CWOUT,
            BCIN, BCOUT, Ci[l], Co[l], res, Hout[l], reluFlag[l]);

        hin = Hout[l];
        K = H_DIM;
    }

    // ---- normalize (in place in d_out emb region), decode, log_softmax
    l2norm_kernel<<<nBlkNW, blk, 0, stream>>>(emb);
    gemm_wmma_f32_kernel<<<dim3(gemmBlocks, 2), blk, 0, stream>>>(emb, dec_W, dec_b, logits, H_DIM, NCLS);
    logsoftmax_kernel<<<N_NODES, blk, 0, stream>>>(logits);
}